// ICSRModule_68676527063204
// MI455X (gfx1250) — compile-verified
//
#include <hip/hip_runtime.h>
#include <hip/hip_bf16.h>

// ---------------------------------------------------------------------------
// Problem constants (match setup_inputs)
// ---------------------------------------------------------------------------
constexpr int BB   = 16;
constexpr int NN   = 4096;
constexpr int SS   = 1024;
constexpr int C1   = 128;
constexpr int C2   = 256;
constexpr int CIN0 = 384;   // C1 + C2
constexpr int M0   = 256;
constexpr int M1   = 256;
constexpr float BN_EPS     = 1e-5f;
constexpr float INTERP_EPS = 1e-8f;

typedef float v2f __attribute__((ext_vector_type(2)));
typedef float v8f __attribute__((ext_vector_type(8)));

// ---------------------------------------------------------------------------
// CDNA5 async memory->LDS helpers (ASYNCcnt path), guarded so compile is safe
// ---------------------------------------------------------------------------
#if __has_builtin(__builtin_amdgcn_global_load_async_to_lds_b128)
#define HAVE_ASYNC_LDS 1
#else
#define HAVE_ASYNC_LDS 0
#endif

#if HAVE_ASYNC_LDS
typedef int v4i_vs __attribute__((vector_size(16)));
typedef __attribute__((address_space(1))) v4i_vs gv4i_t;   // global (prints as __device__)
typedef __attribute__((address_space(3))) v4i_vs lv4i_t;   // LDS    (prints as __shared__)
#endif

__device__ __forceinline__ void lds_async_b128(void* lds, const void* g) {
#if HAVE_ASYNC_LDS
    __builtin_amdgcn_global_load_async_to_lds_b128(
        (gv4i_t*)g, (lv4i_t*)lds, 0, 0);
#else
    *(float4*)lds = *(const float4*)g;
#endif
}

__device__ __forceinline__ void wait_async_lds() {
#if __has_builtin(__builtin_amdgcn_s_wait_asynccnt)
    __builtin_amdgcn_s_wait_asynccnt(0);
#elif HAVE_ASYNC_LDS
    asm volatile("s_wait_asynccnt 0x0" ::: "memory");
#endif
}

// ---------------------------------------------------------------------------
// 1) 3-NN search + interpolation weights (exact reference distance formula)
//    grid (N/256, B), block 256
// ---------------------------------------------------------------------------
__global__ void knn_kernel(const float* __restrict__ xyz1,
                           const float* __restrict__ xyz2,
                           int* __restrict__ idx_out,
                           float* __restrict__ w_out) {
    __shared__ float sx[SS], sy[SS], sz[SS];
    const int b   = blockIdx.y;
    const int n   = blockIdx.x * 256 + threadIdx.x;
    const float* x2 = xyz2 + (size_t)b * 3 * SS;
    for (int i = threadIdx.x; i < SS; i += 256) {
        sx[i] = x2[i];
        sy[i] = x2[SS + i];
        sz[i] = x2[2 * SS + i];
    }
    __syncthreads();

    const float* x1 = xyz1 + (size_t)b * 3 * NN;
    const float qx = x1[n], qy = x1[NN + n], qz = x1[2 * NN + n];
    const float qq = qx * qx + qy * qy + qz * qz;

    float d0 = 3.4e38f, d1 = 3.4e38f, d2 = 3.4e38f;
    int   i0 = 0, i1 = 0, i2 = 0;
    for (int s = 0; s < SS; ++s) {
        const float px = sx[s], py = sy[s], pz = sz[s];
        const float dot = qx * px + qy * py + qz * pz;
        const float pp  = px * px + py * py + pz * pz;
        const float d   = -2.0f * dot + qq + pp;   // same formula as reference
        if (d < d2) {
            if (d < d1) {
                d2 = d1; i2 = i1;
                if (d < d0) { d1 = d0; i1 = i0; d0 = d; i0 = s; }
                else        { d1 = d;  i1 = s; }
            } else { d2 = d; i2 = s; }
        }
    }
    const float c0 = fmaxf(d0, 0.0f), c1 = fmaxf(d1, 0.0f), c2 = fmaxf(d2, 0.0f);
    const float r0 = 1.0f / (c0 + INTERP_EPS);
    const float r1 = 1.0f / (c1 + INTERP_EPS);
    const float r2 = 1.0f / (c2 + INTERP_EPS);
    const float rs = 1.0f / (r0 + r1 + r2);

    const size_t base = ((size_t)b * NN + n) * 3;
    idx_out[base + 0] = i0; idx_out[base + 1] = i1; idx_out[base + 2] = i2;
    w_out[base + 0] = r0 * rs; w_out[base + 1] = r1 * rs; w_out[base + 2] = r2 * rs;
}

// ---------------------------------------------------------------------------
// 2) Build X0 = concat(points1, interp(points2))  -> [B, 384, N]
//    grid (N/256, 384, B), block 256
// ---------------------------------------------------------------------------
__global__ void build_x0_kernel(const float* __restrict__ points1,
                                const float* __restrict__ points2,
                                const int* __restrict__ idx_in,
                                const float* __restrict__ w_in,
                                float* __restrict__ X0) {
    const int b = blockIdx.z;
    const int c = blockIdx.y;
    const int l = blockIdx.x * 256 + threadIdx.x;
    float v;
    if (c < C1) {
        v = points1[((size_t)b * C1 + c) * NN + l];
    } else {
        const int c2 = c - C1;
        const size_t kb = ((size_t)b * NN + l) * 3;
        const int   i0 = idx_in[kb + 0], i1 = idx_in[kb + 1], i2 = idx_in[kb + 2];
        const float w0 = w_in[kb + 0],  w1 = w_in[kb + 1],  w2 = w_in[kb + 2];
        const float* row = points2 + ((size_t)b * C2 + c2) * SS;
        v = w0 * row[i0] + w1 * row[i1] + w2 * row[i2];
    }
    X0[((size_t)b * CIN0 + c) * NN + l] = v;
}

// ---------------------------------------------------------------------------
// 3) Depthwise conv (k=3, SAME, cross-correlation) + per-channel sum/sumsq
//    grid (N/256, C, B), block 256
// ---------------------------------------------------------------------------
template <int C>
__global__ void dwconv_stats_kernel(const float* __restrict__ X,
                                    const float* __restrict__ dw,
                                    float* __restrict__ Y,
                                    float* __restrict__ sum,
                                    float* __restrict__ ssq) {
    const int b = blockIdx.z;
    const int c = blockIdx.y;
    const int l = blockIdx.x * 256 + threadIdx.x;
    const float* xr = X + ((size_t)b * C + c) * NN;
    const float w0 = dw[c * 3 + 0], w1 = dw[c * 3 + 1], w2 = dw[c * 3 + 2];
    const float xm = (l > 0)      ? xr[l - 1] : 0.0f;
    const float xc = xr[l];
    const float xp = (l < NN - 1) ? xr[l + 1] : 0.0f;
    const float y  = w0 * xm + w1 * xc + w2 * xp;
    Y[((size_t)b * C + c) * NN + l] = y;

    __shared__ float rs[256], rq[256];
    rs[threadIdx.x] = y;
    rq[threadIdx.x] = y * y;
    __syncthreads();
    for (int off = 128; off > 0; off >>= 1) {
        if (threadIdx.x < off) {
            rs[threadIdx.x] += rs[threadIdx.x + off];
            rq[threadIdx.x] += rq[threadIdx.x + off];
        }
        __syncthreads();
    }
    if (threadIdx.x == 0) {
        atomicAdd(&sum[c], rs[0]);
        atomicAdd(&ssq[c], rq[0]);
    }
}

// ---------------------------------------------------------------------------
// 4) Finalize BN: fold mean/var + gamma/beta into scale/shift. <<<1, C>>>
// ---------------------------------------------------------------------------
__global__ void finalize_stats_kernel(const float* __restrict__ sum,
                                      const float* __restrict__ ssq,
                                      const float* __restrict__ gamma,
                                      const float* __restrict__ beta,
                                      float* __restrict__ scale,
                                      float* __restrict__ shift,
                                      float invBL) {
    const int c = threadIdx.x;
    const float mean = sum[c] * invBL;
    const float var  = ssq[c] * invBL - mean * mean;
    const float rstd = rsqrtf(var + BN_EPS);
    const float sc   = rstd * gamma[c];
    scale[c] = sc;
    shift[c] = beta[c] - mean * sc;
}

// ---------------------------------------------------------------------------
// 5) Fold channel shuffle (groups=2) into pointwise weights:
//    pw'[o,c] = pw[o, (c % (Cin/2))*2 + c/(Cin/2)]
// ---------------------------------------------------------------------------
__global__ void permute_pw_kernel(const float* __restrict__ pw,
                                  float* __restrict__ pwp,
                                  int Cout, int Cin) {
    const int t = blockIdx.x * 256 + threadIdx.x;
    if (t >= Cout * Cin) return;
    const int o = t / Cin, c = t % Cin;
    const int half = Cin >> 1;
    const int i = (c % half) * 2 + c / half;
    pwp[o * Cin + c] = pw[o * Cin + i];
}

__global__ void zero_f32_kernel(float* __restrict__ p, int n) {
    for (int i = threadIdx.x; i < n; i += 256) p[i] = 0.0f;
}

// ---------------------------------------------------------------------------
// 6) WMMA fp32 GEMM:  Out[b] = Wp (Cout x CIN) * relu(scale*Y[b]+shift) + bias
//    block = 256 threads = 8 waves in 4(o) x 2(l)
//    block tile 128(o) x 64(l); each wave: 32x32 = four 16x16 accumulators
//    Double-buffered LDS; A (weights) tile streamed via async memory->LDS,
//    B (activation) tile via VGPRs with fused BN scale/shift + ReLU.
//    grid (L/64, Cout/128, B)
// ---------------------------------------------------------------------------
template <int CIN>
__global__ void gemm_wmma_kernel(const float* __restrict__ Y,
                                 const float* __restrict__ Wp,
                                 const float* __restrict__ scale,
                                 const float* __restrict__ shift,
                                 const float* __restrict__ bias,
                                 float* __restrict__ Out) {
    // 36-float row stride: 144B (16B aligned for b128) and 36*m mod 64 is a
    // permutation -> conflict-free A-fragment reads.
    __shared__ __align__(16) float As[2][128][36];
    // 80-float row stride: 2*80 mod 64 == 32 -> the two K-half lane groups hit
    // disjoint bank ranges on B-fragment reads; 320B rows keep b128 alignment.
    __shared__ __align__(16) float Bs[2][32][80];

    const int b    = blockIdx.z;
    const int o0   = blockIdx.y * 128;
    const int l0   = blockIdx.x * 64;
    const int tid  = threadIdx.x;
    const int lane = tid & 31;
    const int wave = tid >> 5;
    const int wo   = wave >> 1;    // 0..3 : 32-row band
    const int wl   = wave & 1;     // 0..1 : 32-col band

    const float* Yb = Y + (size_t)b * CIN * NN;

    v8f acc[2][2] = {};

    // --- tile staging helpers (all-uniform control flow) ---
    auto issueA = [&](int k0, int buf) {
        // 128x32 floats = 1024 float4 transfers, 4 per thread, async -> LDS
        #pragma unroll
        for (int i = 0; i < 4; ++i) {
            const int e   = i * 256 + tid;
            const int r   = e >> 3;        // 8 float4 per 32-col row
            const int c4  = e & 7;
            lds_async_b128(&As[buf][r][c4 * 4],
                           &Wp[(size_t)(o0 + r) * CIN + k0 + c4 * 4]);
        }
    };
    auto loadB = [&](int k0, float4* br) {
        // 32x64 floats = 512 float4, 2 per thread
        #pragma unroll
        for (int i = 0; i < 2; ++i) {
            const int e  = i * 256 + tid;
            const int r  = e >> 4;         // 16 float4 per 64-col row
            const int c4 = e & 15;
            br[i] = *(const float4*)&Yb[(size_t)(k0 + r) * NN + l0 + c4 * 4];
        }
    };
    auto storeB = [&](int k0, int buf, const float4* br) {
        #pragma unroll
        for (int i = 0; i < 2; ++i) {
            const int e  = i * 256 + tid;
            const int r  = e >> 4;
            const int c4 = e & 15;
            const int c  = k0 + r;
            const float sc = scale[c], sh = shift[c];
            float4 v = br[i];
            v.x = fmaxf(v.x * sc + sh, 0.0f);
            v.y = fmaxf(v.y * sc + sh, 0.0f);
            v.z = fmaxf(v.z * sc + sh, 0.0f);
            v.w = fmaxf(v.w * sc + sh, 0.0f);
            *(float4*)&Bs[buf][r][c4 * 4] = v;
        }
    };

    // --- prologue: stage first K-slab ---
    {
        float4 br[2];
        issueA(0, 0);
        loadB(0, br);
        storeB(0, 0, br);
    }
    wait_async_lds();
    __syncthreads();

    const int m  = lane & 15;
    const int kh = lane >> 4;     // K-half select (ISA 16x4 fp32 layout)

    int buf = 0;
    for (int k0 = 0; k0 < CIN; k0 += 32) {
        const bool more = (k0 + 32) < CIN;
        float4 br[2];
        if (more) {                       // uniform branch
            issueA(k0 + 32, buf ^ 1);     // async: overlaps the math below
            loadB(k0 + 32, br);           // in-flight global loads
        }

        // 32 WMMAs on current buffers: 4 independent accumulator chains
        #pragma unroll
        for (int kk = 0; kk < 32; kk += 4) {
            v2f a0, a1, b0, b1;
            a0.x = As[buf][wo * 32 +      m][kk + 2 * kh + 0];
            a0.y = As[buf][wo * 32 +      m][kk + 2 * kh + 1];
            a1.x = As[buf][wo * 32 + 16 + m][kk + 2 * kh + 0];
            a1.y = As[buf][wo * 32 + 16 + m][kk + 2 * kh + 1];
            b0.x = Bs[buf][kk + 2 * kh + 0][wl * 32 +      m];
            b0.y = Bs[buf][kk + 2 * kh + 1][wl * 32 +      m];
            b1.x = Bs[buf][kk + 2 * kh + 0][wl * 32 + 16 + m];
            b1.y = Bs[buf][kk + 2 * kh + 1][wl * 32 + 16 + m];
            acc[0][0] = __builtin_amdgcn_wmma_f32_16x16x4_f32(
                false, a0, false, b0, (short)0, acc[0][0], false, false);
            acc[0][1] = __builtin_amdgcn_wmma_f32_16x16x4_f32(
                false, a0, false, b1, (short)0, acc[0][1], false, false);
            acc[1][0] = __builtin_amdgcn_wmma_f32_16x16x4_f32(
                false, a1, false, b0, (short)0, acc[1][0], false, false);
            acc[1][1] = __builtin_amdgcn_wmma_f32_16x16x4_f32(
                false, a1, false, b1, (short)0, acc[1][1], false, false);
        }

        if (more) storeB(k0 + 32, buf ^ 1, br);  // loadcnt waits land here
        wait_async_lds();
        __syncthreads();
        buf ^= 1;
    }

    // C/D layout: VGPR r -> M = r (lanes 0-15) or 8+r (lanes 16-31), N = lane%16
    const int n  = lane & 15;
    const int mh = lane >> 4;
    #pragma unroll
    for (int mt = 0; mt < 2; ++mt) {
        #pragma unroll
        for (int nt = 0; nt < 2; ++nt) {
            const int col = l0 + wl * 32 + nt * 16 + n;
            #pragma unroll
            for (int r = 0; r < 8; ++r) {
                const int orow = o0 + wo * 32 + mt * 16 + mh * 8 + r;
                Out[((size_t)b * 256 + orow) * NN + col] = acc[mt][nt][r] + bias[orow];
            }
        }
    }
}

// ---------------------------------------------------------------------------
// Host-side launcher
// ---------------------------------------------------------------------------
extern "C" void kernel_launch(void* const* d_in, const int* in_sizes, int n_in,
                              void* d_out, int out_size, void* d_ws, size_t ws_size,
                              hipStream_t stream) {
    const float* xyz1    = (const float*)d_in[0];
    const float* xyz2    = (const float*)d_in[1];
    const float* points1 = (const float*)d_in[2];
    const float* points2 = (const float*)d_in[3];
    const float* dw0 = (const float*)d_in[4];
    const float* g0  = (const float*)d_in[5];
    const float* b0  = (const float*)d_in[6];
    const float* pw0 = (const float*)d_in[7];
    const float* pb0 = (const float*)d_in[8];
    const float* dw1 = (const float*)d_in[9];
    const float* g1  = (const float*)d_in[10];
    const float* b1  = (const float*)d_in[11];
    const float* pw1 = (const float*)d_in[12];
    const float* pb1 = (const float*)d_in[13];
    float* out = (float*)d_out;

    char* ws = (char*)d_ws;
    const size_t bytesX  = (size_t)BB * CIN0 * NN * sizeof(float); // 100.7 MB
    const size_t offX    = 0;                      // X0; reused as X1 after GEMM0
    const size_t offY    = offX + bytesX;          // Y0; reused as Y1
    const size_t offIdx  = offY + bytesX;
    const size_t offW    = offIdx + (size_t)BB * NN * 3 * sizeof(int);
    const size_t offSt   = offW + (size_t)BB * NN * 3 * sizeof(float);
    const size_t offPw0  = offSt + (size_t)4 * CIN0 * sizeof(float);
    const size_t offPw1  = offPw0 + (size_t)M0 * CIN0 * sizeof(float);

    float* X     = (float*)(ws + offX);
    float* Y     = (float*)(ws + offY);
    int*   idx   = (int*)(ws + offIdx);
    float* wgt   = (float*)(ws + offW);
    float* stats = (float*)(ws + offSt);
    float* sum   = stats;
    float* ssq   = stats + CIN0;
    float* scale = stats + 2 * CIN0;
    float* shift = stats + 3 * CIN0;
    float* pw0p  = (float*)(ws + offPw0);
    float* pw1p  = (float*)(ws + offPw1);

    const float invBL = 1.0f / (float)(BB * NN);

    // weight shuffle folding
    permute_pw_kernel<<<(M0 * CIN0 + 255) / 256, 256, 0, stream>>>(pw0, pw0p, M0, CIN0);
    permute_pw_kernel<<<(M1 * M0 + 255) / 256, 256, 0, stream>>>(pw1, pw1p, M1, M0);

    // KNN + interpolation weights
    knn_kernel<<<dim3(NN / 256, BB), 256, 0, stream>>>(xyz1, xyz2, idx, wgt);

    // X0 = concat(points1, interp)
    build_x0_kernel<<<dim3(NN / 256, CIN0, BB), 256, 0, stream>>>(points1, points2, idx, wgt, X);

    // ---- ICSR block 0 ----
    zero_f32_kernel<<<1, 256, 0, stream>>>(sum, 2 * CIN0);
    dwconv_stats_kernel<CIN0><<<dim3(NN / 256, CIN0, BB), 256, 0, stream>>>(X, dw0, Y, sum, ssq);
    finalize_stats_kernel<<<1, CIN0, 0, stream>>>(sum, ssq, g0, b0, scale, shift, invBL);
    // GEMM0 writes X1 into the (now dead) X0 buffer
    gemm_wmma_kernel<CIN0><<<dim3(NN / 64, M0 / 128, BB), 256, 0, stream>>>(
        Y, pw0p, scale, shift, pb0, X);

    // ---- ICSR block 1 ----
    zero_f32_kernel<<<1, 256, 0, stream>>>(sum, 2 * M0);
    dwconv_stats_kernel<M0><<<dim3(NN / 256, M0, BB), 256, 0, stream>>>(X, dw1, Y, sum, ssq);
    finalize_stats_kernel<<<1, M0, 0, stream>>>(sum, ssq, g1, b1, scale, shift, invBL);
    gemm_wmma_kernel<M0><<<dim3(NN / 64, M1 / 128, BB), 256, 0, stream>>>(
        Y, pw1p, scale, shift, pb1, out);
}